// WeightedHsLoss_6408091205870
// MI455X (gfx1250) — compile-verified
//
#include <hip/hip_runtime.h>
#include <hip/hip_bf16.h>

typedef __attribute__((ext_vector_type(2))) float v2f;
typedef __attribute__((ext_vector_type(8))) float v8f;

#define ALPHA_C   1.0f
#define BETA_C    1.0f
#define GAMMA_C   1.0f
#define EPS_DENOM 1e-20f
#define B_DIM 128
#define P_DIM 256
#define C_DIM 1000
#define C_TILES 63   // ceil(1000/16)

// Fast HW sqrt: single v_sqrt_f32 (~1 ULP), skips the IEEE refinement expansion.
__device__ __forceinline__ float hw_sqrt(float x) {
    return __builtin_amdgcn_sqrtf(x);
}

// Cross-lane xor-exchange+add via one ds_swizzle_b32 (group-of-32 mode:
// offset[14:10]=xor_mask, offset[9:5]=or_mask=0, offset[4:0]=and_mask=0x1f).
template <int XOR_MASK>
__device__ __forceinline__ float xor_add(float v) {
    int j = __builtin_amdgcn_ds_swizzle(__float_as_int(v), (XOR_MASK << 10) | 0x1f);
    return v + __int_as_float(j);
}

// Workspace layout (floats):
//   [0,256)   colsum[p]   = sum_c W[c,p]^2
//   [256,384) rowsumA[b]  = sum_c sqrt(M[b,c])   (atomic accum)
//   [384,512) sumB[b]     = sum_p |S[b,p]| * sqrt(colsum[p])
//   [512,640) fro_sq[b]   = sum_p S[b,p]^2 * colsum[p]

__global__ void k_colsum(const float* __restrict__ W, float* __restrict__ ws) {
    int p = threadIdx.x;                 // 256 threads, one per column p
    float acc = 0.0f;
    for (int c = 0; c < C_DIM; ++c) {    // coalesced across threads
        float w = W[c * P_DIM + p];
        acc += w * w;
    }
    ws[p] = acc;
    if (p < B_DIM) ws[256 + p] = 0.0f;   // zero rowsumA accumulators
}

__global__ void k_batch(const float* __restrict__ S, const float* __restrict__ ws,
                        float* __restrict__ outB /* = ws+384 */) {
    __shared__ float redF[8], redB[8];
    const int b = blockIdx.x;
    const int p = threadIdx.x;           // 256 threads = 8 waves
    const int lane = p & 31;
    const int wv = p >> 5;
    const float cs = ws[p];
    const float sv = S[b * P_DIM + p];
    float fro_part = sv * sv * cs;
    float b_part   = fabsf(sv) * hw_sqrt(cs);

    // wave32 reduction (full 32 lanes), then 8-wave LDS combine
    fro_part = xor_add<1>(fro_part);  b_part = xor_add<1>(b_part);
    fro_part = xor_add<2>(fro_part);  b_part = xor_add<2>(b_part);
    fro_part = xor_add<4>(fro_part);  b_part = xor_add<4>(b_part);
    fro_part = xor_add<8>(fro_part);  b_part = xor_add<8>(b_part);
    fro_part = xor_add<16>(fro_part); b_part = xor_add<16>(b_part);
    if (lane == 0) { redF[wv] = fro_part; redB[wv] = b_part; }
    __syncthreads();
    if (p == 0) {
        float fs = 0.0f, bs = 0.0f;
#pragma unroll
        for (int i = 0; i < 8; ++i) { fs += redF[i]; bs += redB[i]; }
        outB[b]       = bs;   // sumB   -> ws[384+b]
        outB[128 + b] = fs;   // fro_sq -> ws[512+b]
    }
}

// M = s2 @ w2^T via V_WMMA_F32_16X16X4_F32; accumulate sum_c sqrt(M[b,c]).
// OOB columns (c >= 1000) handled branch-free: clamped address + 0/1 mask,
// so the K-loop is pure load->square->wmma with no exec-mask manipulation.
__global__ void __launch_bounds__(256)
k_wmma(const float* __restrict__ S, const float* __restrict__ W,
       float* __restrict__ rowsumA /* = ws+256 */) {
    const int cb   = blockIdx.x;           // column tile 0..62
    const int wave = threadIdx.x >> 5;     // row tile 0..7 (wave32)
    const int lane = threadIdx.x & 31;
    const int half = lane >> 4;            // 0: K=0,1   1: K=2,3
    const int l16  = lane & 15;

    const int row = wave * 16 + l16;       // batch row (A matrix M index)
    const int col = cb * 16 + l16;         // class col (B matrix N index)
    const int colClamped = (col < C_DIM) ? col : (C_DIM - 1);
    const float bmask = (col < C_DIM) ? 1.0f : 0.0f;   // loop-invariant mask

    const float* Sbase = S + row * P_DIM + 2 * half;
    const float* Wbase = W + colClamped * P_DIM + 2 * half;

    v8f acc = {};
#pragma unroll 4
    for (int k = 0; k < P_DIM / 4; ++k) {
        const int p = k * 4;
        const float ax = Sbase[p], ay = Sbase[p + 1];
        const float bx = Wbase[p], by = Wbase[p + 1];
        v2f a, bm;
        a[0]  = ax * ax;          a[1]  = ay * ay;           // s2 pair
        bm[0] = bx * bx * bmask;  bm[1] = by * by * bmask;   // masked w2 pair
        acc = __builtin_amdgcn_wmma_f32_16x16x4_f32(
            /*neg_a=*/false, a, /*neg_b=*/false, bm,
            /*c_mod=*/(short)0, acc, /*reuse_a=*/false, /*reuse_b=*/false);
    }

    // C/D layout: VGPR j, lanes 0-15 -> row j, lanes 16-31 -> row j+8; N = l16.
    const int rowBase = wave * 16 + 8 * half;
#pragma unroll
    for (int j = 0; j < 8; ++j) {
        float v = hw_sqrt(fmaxf(acc[j], 0.0f));
        // sum the 16 columns held by this half-wave (xor masks stay in-half)
        v = xor_add<1>(v);
        v = xor_add<2>(v);
        v = xor_add<4>(v);
        v = xor_add<8>(v);
        if (l16 == 0) atomicAdd(&rowsumA[rowBase + j], v);
    }
}

__global__ void k_final(const float* __restrict__ ws, float* __restrict__ out) {
    __shared__ float red[4];
    const int b = threadIdx.x;             // 128 threads = 4 waves
    const int lane = b & 31;
    const int wv = b >> 5;
    const float ra  = ws[256 + b];
    const float sb  = ws[384 + b];
    const float fsq = ws[512 + b];
    const float inv_pc = 1.0f / hw_sqrt((float)P_DIM * (float)C_DIM);
    float sp = (ALPHA_C * ra * ra + BETA_C * sb * sb) / (fsq + EPS_DENOM);
    sp = sp * inv_pc + GAMMA_C * hw_sqrt(fsq);

    sp = xor_add<1>(sp);
    sp = xor_add<2>(sp);
    sp = xor_add<4>(sp);
    sp = xor_add<8>(sp);
    sp = xor_add<16>(sp);
    if (lane == 0) red[wv] = sp;
    __syncthreads();
    if (b == 0) out[0] = (red[0] + red[1] + red[2] + red[3]) / (float)B_DIM;
}

extern "C" void kernel_launch(void* const* d_in, const int* in_sizes, int n_in,
                              void* d_out, int out_size, void* d_ws, size_t ws_size,
                              hipStream_t stream) {
    const float* W = (const float*)d_in[0];   // (C=1000, P=256)
    const float* S = (const float*)d_in[1];   // (B=128,  P=256)
    float* ws  = (float*)d_ws;
    float* out = (float*)d_out;

    k_colsum<<<1, 256, 0, stream>>>(W, ws);
    k_batch <<<B_DIM, 256, 0, stream>>>(S, ws, ws + 384);
    k_wmma  <<<C_TILES, 256, 0, stream>>>(S, W, ws + 256);
    k_final <<<1, B_DIM, 0, stream>>>(ws, out);
}